// GAT_37477884625102
// MI455X (gfx1250) — compile-verified
//
#include <hip/hip_runtime.h>
#include <math.h>

// ---------------- problem constants ----------------
#define NN    50000
#define EE    800000
#define ETOT  (EE + NN)          // edges + self loops
#define KDIM  128                // input feature dim for every layer

typedef float v2f __attribute__((ext_vector_type(2)));
typedef float v8f __attribute__((ext_vector_type(8)));

// ---------------- helpers ----------------
__device__ __forceinline__ void atomicMaxF(float* addr, float val) {
    // monotone int encoding: works for mixed-sign floats, init = -inf
    if (val >= 0.0f) atomicMax((int*)addr, __float_as_int(val));
    else             atomicMin((unsigned int*)addr, __float_as_uint(val));
}

// ---------------- mean of edge_weight (1 block) ----------------
__global__ __launch_bounds__(1024) void mean_kernel(const float* __restrict__ ew,
                                                    float* __restrict__ meanw, int E) {
    __shared__ float sdata[1024];
    float s = 0.0f;
    for (int i = threadIdx.x; i < E; i += 1024) s += ew[i];
    sdata[threadIdx.x] = s;
    __syncthreads();
    for (int o = 512; o > 0; o >>= 1) {
        if (threadIdx.x < o) sdata[threadIdx.x] += sdata[threadIdx.x + o];
        __syncthreads();
    }
    if (threadIdx.x == 0) meanw[0] = sdata[0] / (float)E;
}

// ---------------- c[h] = dot(We[h,:], a_e[h,:]) ----------------
__global__ void edge_coef_kernel(const float* __restrict__ We, const float* __restrict__ ae,
                                 float* __restrict__ ce, int H, int F) {
    int h = threadIdx.x;
    if (h >= H) return;
    float s = 0.0f;
    for (int f = 0; f < F; ++f) s += We[h * F + f] * ae[h * F + f];
    ce[h] = s;
}

// ---------------- fp32 WMMA GEMM: out[N,C] = X[N,K] @ W[K,C] ----------------
// One wave per 16x16 output tile.  V_WMMA_F32_16X16X4_F32:
//   A (16x4):  lanes 0-15 hold M=lane, {K=k+0,k+1}; lanes 16-31 hold M=lane-16, {K=k+2,k+3}
//   B (4x16):  lanes 0-15 hold N=lane, rows k+0/k+1; lanes 16-31 hold rows k+2/k+3
//   C/D VGPR v: lanes 0-15 -> M=v,   N=lane; lanes 16-31 -> M=v+8, N=lane-16
__global__ __launch_bounds__(32) void gemm_wmma_f32(const float* __restrict__ X,
                                                    const float* __restrict__ W,
                                                    float* __restrict__ out,
                                                    int K, int C) {
    const int tm   = blockIdx.x;          // M tile
    const int tc   = blockIdx.y;          // C tile
    const int lane = threadIdx.x;
    const int half = lane >> 4;           // 0 or 1
    const int l15  = lane & 15;

    v8f acc = {};
    const float* xrow = X + (size_t)(tm * 16 + l15) * K;
    const int    coln = tc * 16 + l15;

    for (int k = 0; k < K; k += 4) {
        const int kk = k + half * 2;
        v2f a, b;
        a[0] = xrow[kk];
        a[1] = xrow[kk + 1];
        b[0] = W[(size_t)kk * C + coln];
        b[1] = W[(size_t)(kk + 1) * C + coln];
        acc = __builtin_amdgcn_wmma_f32_16x16x4_f32(
            /*neg_a=*/false, a, /*neg_b=*/false, b,
            /*c_mod=*/(short)0, acc, /*reuse_a=*/false, /*reuse_b=*/false);
    }
#pragma unroll
    for (int v = 0; v < 8; ++v) {
        const int row = tm * 16 + v + half * 8;
        out[(size_t)row * C + coln] = acc[v];
    }
}

// ---------------- per-(node,head) attention scores ----------------
__global__ void score_kernel(const float* __restrict__ xw,
                             const float* __restrict__ a_s, const float* __restrict__ a_d,
                             float* __restrict__ ss, float* __restrict__ sd,
                             int H, int F) {
    int i = blockIdx.x * blockDim.x + threadIdx.x;
    if (i >= NN * H) return;
    const int n = i / H, h = i % H;
    const float* v  = xw + (size_t)n * H * F + h * F;
    const float* as = a_s + h * F;
    const float* ad = a_d + h * F;
    float s1 = 0.0f, s2 = 0.0f;
    for (int f = 0; f < F; ++f) { float x = v[f]; s1 += x * as[f]; s2 += x * ad[f]; }
    ss[i] = s1;
    sd[i] = s2;
}

// ---------------- init: out = bias, m = -inf, denom = 0 ----------------
__global__ void init_kernel(float* __restrict__ out, const float* __restrict__ b,
                            float* __restrict__ m, float* __restrict__ denom,
                            int C, int H) {
    const int n = blockIdx.x, t = threadIdx.x;
    out[(size_t)n * C + t] = b[t];
    if (t < H) { m[n * H + t] = -__builtin_inff(); denom[n * H + t] = 0.0f; }
}

// ---------------- pass 1: alpha = leakyrelu(...), segment max ----------------
__global__ void alpha_max_kernel(const int* __restrict__ ei, const float* __restrict__ ew,
                                 const float* __restrict__ meanw,
                                 const float* __restrict__ ss, const float* __restrict__ sd,
                                 const float* __restrict__ ce,
                                 float* __restrict__ alpha, float* __restrict__ m, int H) {
    int idx = blockIdx.x * blockDim.x + threadIdx.x;
    if (idx >= ETOT * H) return;
    const int e = idx / H, h = idx % H;
    int s, d; float ea;
    if (e < EE) { s = ei[e]; d = ei[EE + e]; ea = ew[e]; }
    else        { s = d = e - EE;            ea = meanw[0]; }
    float a = ss[s * H + h] + sd[d * H + h] + ea * ce[h];
    a = a > 0.0f ? a : 0.2f * a;             // leaky relu
    alpha[idx] = a;
    atomicMaxF(&m[d * H + h], a);
}

// ---------------- pass 2: ex = exp(alpha - m[dst]), segment sum ----------------
__global__ void exp_denom_kernel(const int* __restrict__ ei,
                                 const float* __restrict__ m,
                                 float* __restrict__ alpha, float* __restrict__ denom, int H) {
    int idx = blockIdx.x * blockDim.x + threadIdx.x;
    if (idx >= ETOT * H) return;
    const int e = idx / H, h = idx % H;
    const int d = (e < EE) ? ei[EE + e] : (e - EE);
    float ex = __expf(alpha[idx] - m[d * H + h]);
    alpha[idx] = ex;
    atomicAdd(&denom[d * H + h], ex);
}

// ---------------- pass 3: out[dst] += attn * xw[src] (1 block / edge) ----------
__global__ void scatter_kernel(const int* __restrict__ ei,
                               const float* __restrict__ xw,
                               const float* __restrict__ alpha,
                               const float* __restrict__ denom,
                               float* __restrict__ out, int H, int F) {
    const int e = blockIdx.x;
    const int t = threadIdx.x;               // t in [0, H*F)
    int s, d;
    if (e < EE) { s = ei[e]; d = ei[EE + e]; } else { s = d = e - EE; }
    const int h = t / F;
    const float attn = alpha[(size_t)e * H + h] / (denom[d * H + h] + 1e-16f);
    atomicAdd(&out[(size_t)d * H * F + t], xw[(size_t)s * H * F + t] * attn);
}

// ---------------- ELU (layers 0,1) ----------------
__global__ void elu_kernel(float* __restrict__ buf, int n) {
    int i = blockIdx.x * blockDim.x + threadIdx.x;
    if (i >= n) return;
    float v = buf[i];
    buf[i] = v > 0.0f ? v : expm1f(v);
}

// ---------------- host orchestration ----------------
extern "C" void kernel_launch(void* const* d_in, const int* in_sizes, int n_in,
                              void* d_out, int out_size, void* d_ws, size_t ws_size,
                              hipStream_t stream) {
    const float* x  = (const float*)d_in[0];
    const int*   ei = (const int*)d_in[1];      // [2,E] row-major: src then dst
    const float* ew = (const float*)d_in[2];

    const float* Wl[3] = { (const float*)d_in[3],  (const float*)d_in[9],  (const float*)d_in[15] };
    const float* As[3] = { (const float*)d_in[4],  (const float*)d_in[10], (const float*)d_in[16] };
    const float* Ad[3] = { (const float*)d_in[5],  (const float*)d_in[11], (const float*)d_in[17] };
    const float* We[3] = { (const float*)d_in[6],  (const float*)d_in[12], (const float*)d_in[18] };
    const float* Ae[3] = { (const float*)d_in[7],  (const float*)d_in[13], (const float*)d_in[19] };
    const float* Bi[3] = { (const float*)d_in[8],  (const float*)d_in[14], (const float*)d_in[20] };

    const int Hs[3] = { 4, 4, 1 };
    const int Fs[3] = { 32, 32, 64 };           // C = H*F: 128,128,64

    // workspace layout (floats)
    float* ws    = (float*)d_ws;
    float* xw    = ws;                          // NN*128
    float* bufA  = xw    + (size_t)NN * 128;    // NN*128
    float* bufB  = bufA  + (size_t)NN * 128;    // NN*128
    float* ss    = bufB  + (size_t)NN * 128;    // NN*4
    float* sd    = ss    + (size_t)NN * 4;      // NN*4
    float* mseg  = sd    + (size_t)NN * 4;      // NN*4
    float* denom = mseg  + (size_t)NN * 4;      // NN*4
    float* alpha = denom + (size_t)NN * 4;      // ETOT*4
    float* ce    = alpha + (size_t)ETOT * 4;    // 4
    float* meanw = ce + 4;                      // 1

    mean_kernel<<<1, 1024, 0, stream>>>(ew, meanw, EE);

    const float* in = x;
    float* outs[3] = { bufA, bufB, (float*)d_out };

    for (int l = 0; l < 3; ++l) {
        const int H = Hs[l], F = Fs[l], C = H * F;

        edge_coef_kernel<<<1, 32, 0, stream>>>(We[l], Ae[l], ce, H, F);

        // xw = in @ W  (fp32 WMMA, one wave per 16x16 tile)
        gemm_wmma_f32<<<dim3(NN / 16, C / 16), 32, 0, stream>>>(in, Wl[l], xw, KDIM, C);

        score_kernel<<<(NN * H + 255) / 256, 256, 0, stream>>>(xw, As[l], Ad[l], ss, sd, H, F);

        init_kernel<<<NN, C, 0, stream>>>(outs[l], Bi[l], mseg, denom, C, H);

        alpha_max_kernel<<<(ETOT * H + 255) / 256, 256, 0, stream>>>(
            ei, ew, meanw, ss, sd, ce, alpha, mseg, H);

        exp_denom_kernel<<<(ETOT * H + 255) / 256, 256, 0, stream>>>(ei, mseg, alpha, denom, H);

        scatter_kernel<<<ETOT, C, 0, stream>>>(ei, xw, alpha, denom, outs[l], H, F);

        if (l < 2) elu_kernel<<<((NN * C) + 255) / 256, 256, 0, stream>>>(outs[l], NN * C);

        in = outs[l];
    }
}